// EikonalLoss3d_88124138979534
// MI455X (gfx1250) — compile-verified
//
#include <hip/hip_runtime.h>
#include <cstdint>
#include <cstddef>

// ---------------------------------------------------------------------------
// Eikonal loss, (2,1,256,256,256) fp32 -> scalar mean(| |grad| - 1 |).
// Bandwidth-bound: 134 MB input, ~5.8us floor at 23.3 TB/s (amp ~1.13 here).
// CDNA5 paths: global_load_async_to_lds_b128/b32 + s_wait_asynccnt streaming
// of D-planes through a 3-deep LDS ring; v_wmma_f32_16x16x4_f32 ones-matrix
// wave reduction; deterministic block partials in d_ws + finalize kernel.
// ---------------------------------------------------------------------------

typedef __attribute__((ext_vector_type(2))) float v2f;
typedef __attribute__((ext_vector_type(8))) float v8f;

#define TH 16        // output rows (H) per block tile
#define TD 16        // output planes (D) per block chunk
#define ROWS 17      // rows resident per plane tile (TH + 1 halo)
#define W256 256
#define PLANE_F (256 * 256)          // floats per (d) plane
#define N_OUT_TOTAL 33162750.0f      // 2 * 255^3

// Wave-level sum of one float per lane using V_WMMA_F32_16X16X4_F32 with an
// all-ones B matrix.  A layout (16x4 fp32): lanes 0-15 -> {K0,K1}, lanes
// 16-31 -> {K2,K3}; with a = (v, 0) every D row m equals v[m] + v[m+16].
// Lanes 0-15 hold rows 0-7 of D, lanes 16-31 hold rows 8-15, so one
// shfl_xor(16) completes the 32-lane sum (valid on every lane).
__device__ __forceinline__ float wave_reduce_sum(float v) {
#if defined(__gfx1250__) && __has_builtin(__builtin_amdgcn_wmma_f32_16x16x4_f32)
  v2f a; a.x = v;    a.y = 0.0f;
  v2f b; b.x = 1.0f; b.y = 1.0f;
  v8f c = {0.f, 0.f, 0.f, 0.f, 0.f, 0.f, 0.f, 0.f};
  v8f d = __builtin_amdgcn_wmma_f32_16x16x4_f32(false, a, false, b,
                                                (short)0, c, false, false);
  float s = ((d[0] + d[1]) + (d[2] + d[3])) + ((d[4] + d[5]) + (d[6] + d[7]));
  s += __shfl_xor(s, 16, 32);
  return s;
#else
  for (int off = 16; off > 0; off >>= 1) v += __shfl_xor(v, off, 32);
  return v;
#endif
}

__device__ __forceinline__ float fast_sqrtf(float x) {
#if defined(__gfx1250__) && __has_builtin(__builtin_amdgcn_sqrtf)
  return __builtin_amdgcn_sqrtf(x);   // raw v_sqrt_f32, no IEEE fixup chain
#else
  return sqrtf(x);
#endif
}

// Load one 17-row x 1KB plane tile into LDS with exactly 5 uniform async ops
// per wave: 4x B128 sweeps (4 rows each: 64 lanes x 16B per row) + 1x B32
// sweep for the halo row (256 lanes x 4B = 1 row).  Every wave is fully
// active in every sweep, so ASYNCcnt accounting is identical block-wide.
__device__ __forceinline__ void async_load_plane(uint32_t ldsBase,
                                                 const float* __restrict__ plane,
                                                 int h0, int tid) {
#pragma unroll
  for (int s = 0; s < 4; ++s) {
    int rloc = s * 4 + (tid >> 6);                    // local row 0..15
    int rowg = h0 + rloc;
    rowg = (rowg > 255) ? 255 : rowg;                 // clamp at H edge
    uint32_t gOff = (uint32_t)(rowg * 1024 + (tid & 63) * 16);
    uint32_t lOff = ldsBase + (uint32_t)(rloc * 1024 + (tid & 63) * 16);
    asm volatile("global_load_async_to_lds_b128 %0, %1, %2"
                 :
                 : "v"(lOff), "v"(gOff), "s"(plane)
                 : "memory");
  }
  {
    int rowg = h0 + 16;
    rowg = (rowg > 255) ? 255 : rowg;
    uint32_t gOff = (uint32_t)(rowg * 1024 + tid * 4);
    uint32_t lOff = ldsBase + (uint32_t)(16 * 1024 + tid * 4);
    asm volatile("global_load_async_to_lds_b32 %0, %1, %2"
                 :
                 : "v"(lOff), "v"(gOff), "s"(plane)
                 : "memory");
  }
}

__global__ __launch_bounds__(256) void EikonalLoss3d_main(
    const float* __restrict__ in, float* __restrict__ ws) {
  __shared__ float ring[3][ROWS * W256];   // 3 x 17 KB plane tiles
  __shared__ float wsum[8];

  const int tid    = threadIdx.x;
  const int blk    = blockIdx.x;           // 512 blocks total
  const int bIdx   = blk >> 8;             // batch (0..1)
  const int rem    = blk & 255;
  const int hTile  = rem & 15;             // 16 H tiles of 16 rows
  const int dChunk = rem >> 4;             // 16 D chunks of 16 planes
  const int h0 = hTile * TH;
  const int d0 = dChunk * TD;

  const int nOut     = (255 - d0) < TD ? (255 - d0) : TD;   // output planes
  const int nRowsOut = (255 - h0) < TH ? (255 - h0) : TH;   // output rows
  const int numPlanes = nOut + 1;

  const float* base = in + ((size_t)bIdx << 24);   // b * 256^3 floats

  uint32_t lbs[3];
  lbs[0] = (uint32_t)(size_t)(void*)(&ring[0][0]);
  lbs[1] = (uint32_t)(size_t)(void*)(&ring[1][0]);
  lbs[2] = (uint32_t)(size_t)(void*)(&ring[2][0]);

  // Prologue: planes d0 and d0+1 into ring slots 0,1 (10 outstanding ops).
  async_load_plane(lbs[0], base + (size_t)d0 * PLANE_F, h0, tid);
  async_load_plane(lbs[1], base + (size_t)(d0 + 1) * PLANE_F, h0, tid);

  float acc = 0.0f;

  for (int i = 0; i < nOut; ++i) {
    __syncthreads();   // readers of the slot we are about to overwrite are done

    if (i + 2 < numPlanes) {
      // Prefetch plane d0+i+2, then wait until only those 5 new ops remain:
      // async ops complete in order, so planes i and i+1 are then resident.
      async_load_plane(lbs[(i + 2) % 3], base + (size_t)(d0 + i + 2) * PLANE_F,
                       h0, tid);
      asm volatile("s_wait_asynccnt 5" ::: "memory");
    } else {
      asm volatile("s_wait_asynccnt 0" ::: "memory");
    }
    __syncthreads();   // all waves' async data visible block-wide

    const float* __restrict__ cur = &ring[i % 3][0];        // plane d
    const float* __restrict__ nxt = &ring[(i + 1) % 3][0];  // plane d+1

    if (tid < 255) {
      const int w = tid;
      const float sc = 256.0f;
#pragma unroll 4
      for (int rr = 0; rr < nRowsOut; ++rr) {
        float core = cur[rr * W256 + w];
        float xw   = cur[rr * W256 + w + 1];
        float xh   = cur[(rr + 1) * W256 + w];
        float xd   = nxt[rr * W256 + w];
        float dd = (core - xd) * sc;
        float dh = (core - xh) * sc;
        float dw = (core - xw) * sc;
        float nrm = fast_sqrtf(fmaf(dd, dd, fmaf(dh, dh, dw * dw)));
        acc += fabsf(nrm - 1.0f);
      }
    }
  }

  // Block reduction: WMMA wave sums -> 8 partials -> one float per block.
  float wtot = wave_reduce_sum(acc);       // EXEC all-ones here (no divergence)
  const int wave = tid >> 5;
  const int lane = tid & 31;
  if (lane == 0) wsum[wave] = wtot;
  __syncthreads();
  if (tid == 0) {
    float bt = 0.0f;
#pragma unroll
    for (int k = 0; k < 8; ++k) bt += wsum[k];
    ws[blk] = bt;                          // deterministic: one slot per block
  }
}

__global__ __launch_bounds__(256) void EikonalLoss3d_finalize(
    const float* __restrict__ ws, float* __restrict__ out) {
  __shared__ float block_part[8];
  const int tid = threadIdx.x;
  // 512 partials, 256 threads: two strided reads keep the add order fixed.
  float v = ws[tid] + ws[tid + 256];
  float t = wave_reduce_sum(v);
  if ((tid & 31) == 0) block_part[tid >> 5] = t;
  __syncthreads();
  if (tid == 0) {
    float s = 0.0f;
#pragma unroll
    for (int k = 0; k < 8; ++k) s += block_part[k];
    out[0] = s * (1.0f / N_OUT_TOTAL);
  }
}

extern "C" void kernel_launch(void* const* d_in, const int* in_sizes, int n_in,
                              void* d_out, int out_size, void* d_ws, size_t ws_size,
                              hipStream_t stream) {
  (void)in_sizes; (void)n_in; (void)out_size; (void)ws_size;
  const float* in = (const float*)d_in[0];
  float* out = (float*)d_out;
  float* ws  = (float*)d_ws;               // needs 512 floats (2 KB)

  EikonalLoss3d_main<<<512, 256, 0, stream>>>(in, ws);
  EikonalLoss3d_finalize<<<1, 256, 0, stream>>>(ws, out);
}